// GCN_2954937500451
// MI455X (gfx1250) — compile-verified
//
#include <hip/hip_runtime.h>
#include <hip/hip_bf16.h>

#define DFEAT 128
#define NEG_SLOPE 0.01f
#define LN_EPS 1e-5f

typedef __attribute__((ext_vector_type(2))) float v2f;
typedef __attribute__((ext_vector_type(8))) float v8f;

// ---------------------------------------------------------------- utilities
__device__ __forceinline__ float leaky(float v) {
    return v > 0.0f ? v : NEG_SLOPE * v;
}

__device__ __forceinline__ float waveReduceSum(float v) {
#pragma unroll
    for (int off = 16; off > 0; off >>= 1)
        v += __shfl_xor(v, off, 32);
    return v;
}

// -------------------------------------------------------------- degree pass
__global__ void gcn_deg_init(float* __restrict__ deg, int n) {
    int i = blockIdx.x * blockDim.x + threadIdx.x;
    if (i < n) deg[i] = 1.0f;                    // self-loop contribution
}

__global__ void gcn_deg_count(const long long* __restrict__ dst,
                              float* __restrict__ deg, int e) {
    int i = blockIdx.x * blockDim.x + threadIdx.x;
    if (i < e) atomicAdd(&deg[(int)dst[i]], 1.0f);
}

__global__ void gcn_deg_rsqrt(float* __restrict__ deg, int n) {
    int i = blockIdx.x * blockDim.x + threadIdx.x;
    if (i < n) deg[i] = rsqrtf(deg[i]);          // deg >= 1 always
}

__global__ void gcn_zero(float* __restrict__ p, size_t n) {
    size_t i = (size_t)blockIdx.x * blockDim.x + threadIdx.x;
    if (i < n) p[i] = 0.0f;
}

// ------------------------------------------------------- fp32 WMMA GEMM
// out[N x 128] = A[N x 128] @ W[128 x 128]
// One workgroup (8 waves) per 16-row stripe; wave w owns columns [16w,16w+16).
// A stripe staged in LDS (row pitch 132 floats -> 16 rows land in 16 banks).
__global__ __launch_bounds__(256)
void gcn_gemm_wmma(const float* __restrict__ A,
                   const float* __restrict__ W,
                   float* __restrict__ out, int nrows) {
    __shared__ float sA[16][132];

    const int tid  = threadIdx.x;
    const int wave = tid >> 5;
    const int lane = tid & 31;
    const int row0 = blockIdx.x * 16;

    // cooperative stage of the 16x128 A stripe
#pragma unroll
    for (int i = tid; i < 16 * DFEAT; i += 256) {
        int r = i >> 7, c = i & 127;
        sA[r][c] = A[(size_t)(row0 + r) * DFEAT + c];
    }
    __syncthreads();

    const int col0 = wave * 16;
    const int m    = lane & 15;            // A row within tile
    const int n    = lane & 15;            // B/D column within tile
    const int kb   = (lane >> 4) * 2;      // lanes 16-31 hold K+2,K+3

    v8f acc = {0.f, 0.f, 0.f, 0.f, 0.f, 0.f, 0.f, 0.f};

#pragma unroll
    for (int k0 = 0; k0 < DFEAT; k0 += 4) {
        v2f a, b;
        a.x = sA[m][k0 + kb];
        a.y = sA[m][k0 + kb + 1];
        b.x = W[(size_t)(k0 + kb)     * DFEAT + col0 + n];
        b.y = W[(size_t)(k0 + kb + 1) * DFEAT + col0 + n];
        acc = __builtin_amdgcn_wmma_f32_16x16x4_f32(
                  false, a, false, b, (short)0, acc, false, false);
    }

    // D layout: VGPR v -> row v (lanes 0-15) / v+8 (lanes 16-31), col = lane&15
    const int mbase = row0 + (lane >> 4) * 8;
    const int ocol  = col0 + n;
#pragma unroll
    for (int v = 0; v < 8; ++v)
        out[(size_t)(mbase + v) * DFEAT + ocol] = acc[v];
}

// ------------------------------------------------- edge gather/scatter pass
// One wave32 per edge: 32 lanes x float4 == 128 features.
__global__ __launch_bounds__(256)
void gcn_scatter(const float* __restrict__ h,
                 const long long* __restrict__ src,
                 const long long* __restrict__ dst,
                 const float* __restrict__ dinv,
                 float* __restrict__ agg, int e) {
    int edge = (int)(((size_t)blockIdx.x * blockDim.x + threadIdx.x) >> 5);
    int lane = threadIdx.x & 31;
    if (edge >= e) return;

    int s = (int)src[edge];
    int d = (int)dst[edge];
    float c = dinv[s] * dinv[d];

    float4 v = *(const float4*)(h + (size_t)s * DFEAT + lane * 4);
    float* ap = agg + (size_t)d * DFEAT + lane * 4;
    atomicAdd(ap + 0, v.x * c);
    atomicAdd(ap + 1, v.y * c);
    atomicAdd(ap + 2, v.z * c);
    atomicAdd(ap + 3, v.w * c);
}

// ------------------------------------- fused self-loop + bias + act + LN
// One wave per node row; lane holds 4 consecutive features.
// FINAL==false: y = LN( leaky(agg + c*h + bias) + x ) * g + b
// FINAL==true : y = leaky( LN( (agg + c*h + bias) + x ) * g + b )
template <bool FINAL>
__global__ __launch_bounds__(256)
void gcn_post(const float* __restrict__ agg, const float* __restrict__ h,
              const float* __restrict__ x, const float* __restrict__ bias,
              const float* __restrict__ gamma, const float* __restrict__ beta,
              const float* __restrict__ dinv, float* __restrict__ out, int n) {
    int node = (int)(((size_t)blockIdx.x * blockDim.x + threadIdx.x) >> 5);
    int lane = threadIdx.x & 31;
    if (node >= n) return;

    size_t base = (size_t)node * DFEAT + lane * 4;
    float c = dinv[node];
    c = c * c;                                   // self-loop coefficient

    float4 a  = *(const float4*)(agg + base);
    float4 hv = *(const float4*)(h + base);
    float4 xv = *(const float4*)(x + base);
    float4 bv = *(const float4*)(bias + lane * 4);

    float4 t;
    t.x = a.x + c * hv.x + bv.x;
    t.y = a.y + c * hv.y + bv.y;
    t.z = a.z + c * hv.z + bv.z;
    t.w = a.w + c * hv.w + bv.w;

    if (!FINAL) {
        t.x = leaky(t.x); t.y = leaky(t.y); t.z = leaky(t.z); t.w = leaky(t.w);
    }
    t.x += xv.x; t.y += xv.y; t.z += xv.z; t.w += xv.w;

    float s  = waveReduceSum(t.x + t.y + t.z + t.w);
    float sq = waveReduceSum(t.x * t.x + t.y * t.y + t.z * t.z + t.w * t.w);
    float mu  = s * (1.0f / DFEAT);
    float var = sq * (1.0f / DFEAT) - mu * mu;
    float r   = rsqrtf(var + LN_EPS);

    float4 gv  = *(const float4*)(gamma + lane * 4);
    float4 bev = *(const float4*)(beta + lane * 4);

    float4 o;
    o.x = (t.x - mu) * r * gv.x + bev.x;
    o.y = (t.y - mu) * r * gv.y + bev.y;
    o.z = (t.z - mu) * r * gv.z + bev.z;
    o.w = (t.w - mu) * r * gv.w + bev.w;

    if (FINAL) {
        o.x = leaky(o.x); o.y = leaky(o.y); o.z = leaky(o.z); o.w = leaky(o.w);
    }
    *(float4*)(out + base) = o;
}

// ---------------------------------------------------------------- launcher
extern "C" void kernel_launch(void* const* d_in, const int* in_sizes, int n_in,
                              void* d_out, int out_size, void* d_ws, size_t ws_size,
                              hipStream_t stream) {
    const float*      x   = (const float*)d_in[0];
    const long long*  ei  = (const long long*)d_in[1];   // reference: jnp.int64
    const float*      W1  = (const float*)d_in[2];
    const float*      b1  = (const float*)d_in[3];
    const float*      g1  = (const float*)d_in[4];
    const float*      be1 = (const float*)d_in[5];
    const float*      W2  = (const float*)d_in[6];
    const float*      b2  = (const float*)d_in[7];
    const float*      g2  = (const float*)d_in[8];
    const float*      be2 = (const float*)d_in[9];

    const int N = in_sizes[0] / DFEAT;      // 50000
    const int E = in_sizes[1] / 2;          // 800000
    const long long* src = ei;
    const long long* dst = ei + E;

    const size_t nd = (size_t)N * DFEAT;
    char*  ws   = (char*)d_ws;
    float* dinv = (float*)ws;
    size_t off  = ((size_t)N * sizeof(float) + 255) & ~(size_t)255;
    float* hbuf = (float*)(ws + off);       // N x D (reused by both layers)
    float* aggb = hbuf + nd;                // N x D accumulator
    float* y1   = aggb + nd;                // N x D layer-1 output
    float* out  = (float*)d_out;

    const int T = 256;
    dim3 bN((N + T - 1) / T), bE((E + T - 1) / T);
    dim3 bND((unsigned)((nd + T - 1) / T));
    dim3 bGemm((N + 15) / 16);
    dim3 bEdge((E + 7) / 8);                // 8 waves/block, 1 wave/edge
    dim3 bNode((N + 7) / 8);                // 8 waves/block, 1 wave/node

    // degrees -> dinv
    gcn_deg_init <<<bN, T, 0, stream>>>(dinv, N);
    gcn_deg_count<<<bE, T, 0, stream>>>(dst, dinv, E);
    gcn_deg_rsqrt<<<bN, T, 0, stream>>>(dinv, N);

    // ---- layer 1
    gcn_zero<<<bND, T, 0, stream>>>(aggb, nd);
    gcn_gemm_wmma<<<bGemm, T, 0, stream>>>(x, W1, hbuf, N);
    gcn_scatter<<<bEdge, T, 0, stream>>>(hbuf, src, dst, dinv, aggb, E);
    gcn_post<false><<<bNode, T, 0, stream>>>(aggb, hbuf, x, b1, g1, be1, dinv, y1, N);

    // ---- layer 2 (residual is the ORIGINAL x)
    gcn_zero<<<bND, T, 0, stream>>>(aggb, nd);
    gcn_gemm_wmma<<<bGemm, T, 0, stream>>>(y1, W2, hbuf, N);
    gcn_scatter<<<bEdge, T, 0, stream>>>(hbuf, src, dst, dinv, aggb, E);
    gcn_post<true><<<bNode, T, 0, stream>>>(aggb, hbuf, x, b2, g2, be2, dinv, out, N);
}